// ScaledGraphRELoss_40346922778989
// MI455X (gfx1250) — compile-verified
//
#include <hip/hip_runtime.h>
#include <hip/hip_bf16.h>
#include <math.h>

// ---------------------------------------------------------------------------
// ScaledGraphRELoss for MI455X (gfx1250, wave32).
// Memory-bound streaming reduction (~250-290 MB -> ~12us @ 23.3 TB/s).
// D=4 per-node L1 reduction done with V_WMMA_F32_16X16X4_F32:
//   A = 16x4 tile of |pred-target| (or |target|), B = 4x16 ones,
//   D column j = per-node row sums (full fp32 precision).
// Per-graph bins privatized in LDS (1024 x 2 f32), ds_add_f32 atomics,
// flushed once per block with global_atomic_add_f32.
// Force magnitude uses hardware v_sqrt_f32 (1 ulp) instead of the IEEE
// refinement sequence -- negligible error in a 4M-element loss sum.
// ---------------------------------------------------------------------------

typedef __attribute__((ext_vector_type(2))) float v2f;
typedef __attribute__((ext_vector_type(8))) float v8f;

#define NUM_GRAPHS 1024
#define LOSS_EPS   1e-8f
#define MIN_SCALE  0.1f

__global__ void zero_ws_kernel(float* __restrict__ ws, int n) {
    int i = blockIdx.x * blockDim.x + threadIdx.x;
    if (i < n) ws[i] = 0.0f;
}

// select c[idx] from a v8f without scratch spills (chained cndmask)
__device__ __forceinline__ float sel8(v8f v, int idx) {
    float r = v[0];
    r = (idx == 1) ? v[1] : r;
    r = (idx == 2) ? v[2] : r;
    r = (idx == 3) ? v[3] : r;
    r = (idx == 4) ? v[4] : r;
    r = (idx == 5) ? v[5] : r;
    r = (idx == 6) ? v[6] : r;
    r = (idx == 7) ? v[7] : r;
    return r;
}

__device__ __forceinline__ float fast_hypot2(float fx, float fy) {
    // v_sqrt_f32: ~1 ulp, fine for loss-scale accumulation
    return __builtin_amdgcn_sqrtf(fmaf(fx, fx, fy * fy));
}

__global__ __launch_bounds__(256) void graph_loss_main_kernel(
    const float* __restrict__ pred,     // [N,4]
    const float* __restrict__ target,   // [N,4]
    const int*   __restrict__ batch2,   // int64 batch viewed as int pairs (LE low dword at 2*i)
    const float* __restrict__ x,        // [N,8]
    float* __restrict__ g_err,          // [1024]
    float* __restrict__ g_tgt,          // [1024]
    float* __restrict__ g_force,        // [1]
    int N)
{
    __shared__ float s_err[NUM_GRAPHS];
    __shared__ float s_tgt[NUM_GRAPHS];
    __shared__ float s_red[256];

    const int tid = threadIdx.x;
    for (int i = tid; i < NUM_GRAPHS; i += blockDim.x) { s_err[i] = 0.0f; s_tgt[i] = 0.0f; }
    __syncthreads();

    const int lane = tid & 31;
    const int row  = lane & 15;   // node within 16-node tile
    const int hi   = lane >> 4;   // 0: components 0-1, 1: components 2-3 (A-matrix layout)
    const int wavesPerBlock = blockDim.x >> 5;
    const int globalWave = blockIdx.x * wavesPerBlock + (tid >> 5);
    const int numWaves   = gridDim.x * wavesPerBlock;
    const int numTiles   = N >> 4;

    const v2f bones = {1.0f, 1.0f};   // B = ones: every D column = row sums of A
    float facc = 0.0f;

    for (int tile = globalWave; tile < numTiles; tile += numWaves) {
        const int node = (tile << 4) + row;
        // Wave covers 16 consecutive nodes * 16B = 256B of pred and of target,
        // fully coalesced as 32 lanes x b64 loads.
        const v2f p = ((const v2f*)pred)[node * 2 + hi];
        const v2f t = ((const v2f*)target)[node * 2 + hi];

        v2f ae; ae.x = fabsf(p.x - t.x); ae.y = fabsf(p.y - t.y);
        v2f at; at.x = fabsf(t.x);       at.y = fabsf(t.y);

        v8f z = {};
        // EXEC is all-1s here (wave-uniform control flow around WMMA).
        v8f cerr = __builtin_amdgcn_wmma_f32_16x16x4_f32(
            false, ae, false, bones, (short)0, z, false, false);
        v8f ctgt = __builtin_amdgcn_wmma_f32_16x16x4_f32(
            false, at, false, bones, (short)0, z, false, false);

        // D layout: rows 0-7 live in c[m] on lanes 0-15, rows 8-15 in c[m-8]
        // on lanes 16-31. Lanes 0-7 bin nodes 0-7; lanes 16-23 bin nodes 8-15.
        int myRow = -1;
        if (lane < 8)                     myRow = lane;
        else if (lane >= 16 && lane < 24) myRow = 8 + (lane - 16);
        if (myRow >= 0) {
            const int n2  = (tile << 4) + myRow;
            const int g   = batch2[2 * n2];      // low dword of sorted int64 id (0..1023)
            const int idx = myRow & 7;
            atomicAdd(&s_err[g], sel8(cerr, idx));
            atomicAdd(&s_tgt[g], sel8(ctgt, idx));
        }

        // force_mag = ||x[node,3:5]||, lanes 0-15 each handle one node
        if (hi == 0) {
            const float fx = x[node * 8 + 3];
            const float fy = x[node * 8 + 4];
            facc += fast_hypot2(fx, fy);
        }
    }

    // Scalar tail (N not multiple of 16): handled by first few global threads.
    const int rem = N & 15;
    const int gtid = blockIdx.x * blockDim.x + tid;
    if (gtid < rem) {
        const int node = (numTiles << 4) + gtid;
        float e = 0.0f, tv = 0.0f;
        #pragma unroll
        for (int d = 0; d < 4; ++d) {
            const float pv = pred[node * 4 + d];
            const float tg = target[node * 4 + d];
            e  += fabsf(pv - tg);
            tv += fabsf(tg);
        }
        const int g = batch2[2 * node];
        atomicAdd(&s_err[g], e);
        atomicAdd(&s_tgt[g], tv);
        const float fx = x[node * 8 + 3];
        const float fy = x[node * 8 + 4];
        facc += fast_hypot2(fx, fy);
    }

    __syncthreads();
    // Flush privatized bins (skip untouched bins to cut atomic traffic;
    // a block sees ~few graphs since batch is sorted).
    for (int i = tid; i < NUM_GRAPHS; i += blockDim.x) {
        const float e = s_err[i];
        if (e != 0.0f) atomicAdd(&g_err[i], e);
        const float tv = s_tgt[i];
        if (tv != 0.0f) atomicAdd(&g_tgt[i], tv);
    }

    // Block-reduce the force partial, one global atomic per block.
    s_red[tid] = facc;
    __syncthreads();
    for (int off = blockDim.x >> 1; off > 0; off >>= 1) {
        if (tid < off) s_red[tid] += s_red[tid + off];
        __syncthreads();
    }
    if (tid == 0) atomicAdd(g_force, s_red[0]);
}

__global__ __launch_bounds__(1024) void graph_loss_finalize_kernel(
    const float* __restrict__ g_err,
    const float* __restrict__ g_tgt,
    const float* __restrict__ g_force,
    float* __restrict__ out)
{
    __shared__ float s[NUM_GRAPHS];
    const int g = threadIdx.x;
    s[g] = g_err[g] / (g_tgt[g] + LOSS_EPS);
    __syncthreads();
    for (int off = NUM_GRAPHS >> 1; off > 0; off >>= 1) {
        if (g < off) s[g] += s[g + off];
        __syncthreads();
    }
    if (g == 0) {
        const float scale = fmaxf(g_force[0], MIN_SCALE);
        out[0] = (s[0] / (float)NUM_GRAPHS) * scale * 100.0f;
    }
}

extern "C" void kernel_launch(void* const* d_in, const int* in_sizes, int n_in,
                              void* d_out, int out_size, void* d_ws, size_t ws_size,
                              hipStream_t stream) {
    const float* pred   = (const float*)d_in[0];
    const float* target = (const float*)d_in[1];
    const int*   batch2 = (const int*)d_in[2];   // int64 -> LE low dwords at even idx
    const float* x      = (const float*)d_in[3];
    const int N = in_sizes[0] / 4;               // pred is [N,4]

    float* ws      = (float*)d_ws;
    float* g_err   = ws;                  // [1024]
    float* g_tgt   = ws + NUM_GRAPHS;     // [1024]
    float* g_force = ws + 2 * NUM_GRAPHS; // [1]
    const int nzero = 2 * NUM_GRAPHS + 1;

    zero_ws_kernel<<<(nzero + 255) / 256, 256, 0, stream>>>(ws, nzero);

    const int blocks = 1024;   // 8192 waves; 250k tiles -> ~30 tiles/wave
    graph_loss_main_kernel<<<blocks, 256, 0, stream>>>(
        pred, target, batch2, x, g_err, g_tgt, g_force, N);

    graph_loss_finalize_kernel<<<1, 1024, 0, stream>>>(
        g_err, g_tgt, g_force, (float*)d_out);
}